// Aggregator_53523882443255
// MI455X (gfx1250) — compile-verified
//
#include <hip/hip_runtime.h>
#include <hip/hip_bf16.h>

typedef __attribute__((ext_vector_type(2))) float v2f;
typedef __attribute__((ext_vector_type(8))) float v8f;

// One wave handles (node_tile of 16 rows) x (one 16-wide feature tile).
// Segment-sum expressed as one-hot A (16x4 f32) x gathered B (4x16 f32)
// accumulated with V_WMMA_F32_16X16X4_F32 (exact f32 math).
__global__ void __launch_bounds__(256)
seg_sum_wmma_kernel(const float* __restrict__ feat,
                    const int* __restrict__ nidx,
                    const int* __restrict__ seg,
                    float* __restrict__ out,
                    int n_nodes, int n_edges) {
    const int gtid  = blockIdx.x * blockDim.x + threadIdx.x;
    const int wave  = gtid >> 5;
    const int lane  = gtid & 31;

    const int ntile = wave & 3;        // which 16-wide slice of the 64 features
    const int tile  = wave >> 2;       // which group of 16 destination nodes
    const int base  = tile * 16;
    if (base >= n_nodes) return;

    const int m     = lane & 15;       // row (M) for A, column (N) for B/C/D
    const int khalf = lane >> 4;       // 0: K={0,1}, 1: K={2,3}
    const int col0  = ntile * 16;
    const int target = base + m;

    // Wave-uniform binary searches; keep lo/hi in SGPRs so the searched loads
    // have provably uniform addresses (scalar path) and loop control is SALU.
    int lo = 0, hi = n_edges;
    while (lo < hi) {
        int mid = (lo + hi) >> 1;
        int s = __builtin_amdgcn_readfirstlane(seg[mid]);
        if (s < base) lo = mid + 1; else hi = mid;
    }
    const int start = __builtin_amdgcn_readfirstlane(lo);
    hi = n_edges;
    while (lo < hi) {
        int mid = (lo + hi) >> 1;
        int s = __builtin_amdgcn_readfirstlane(seg[mid]);
        if (s < base + 16) lo = mid + 1; else hi = mid;
    }
    const int end = __builtin_amdgcn_readfirstlane(lo);

    v8f c0 = {}, c1 = {};

    int e = start;
    // Main loop: 8 edges (two K=4 chunks) per iteration, no guards, two
    // independent accumulator chains on the matrix pipe.
    for (; e + 8 <= end; e += 8) {
        __builtin_prefetch(seg + e + 128, 0, 1);   // global_prefetch_b8
        __builtin_prefetch(nidx + e + 128, 0, 1);

        const int ea0 = e + khalf * 2;             // chunk 0: K = 0/2
        const int eb0 = ea0 + 1;                   //          K = 1/3
        const int ea1 = ea0 + 4;                   // chunk 1
        const int eb1 = eb0 + 4;

        const int sa0 = seg[ea0], sb0 = seg[eb0];
        const int sa1 = seg[ea1], sb1 = seg[eb1];
        const int ra0 = nidx[ea0], rb0 = nidx[eb0];
        const int ra1 = nidx[ea1], rb1 = nidx[eb1];

        v2f a0; a0.x = (sa0 == target) ? 1.0f : 0.0f;
                a0.y = (sb0 == target) ? 1.0f : 0.0f;
        v2f a1; a1.x = (sa1 == target) ? 1.0f : 0.0f;
                a1.y = (sb1 == target) ? 1.0f : 0.0f;

        v2f b0; b0.x = feat[ra0 * 64 + col0 + m];
                b0.y = feat[rb0 * 64 + col0 + m];
        v2f b1; b1.x = feat[ra1 * 64 + col0 + m];
                b1.y = feat[rb1 * 64 + col0 + m];

        c0 = __builtin_amdgcn_wmma_f32_16x16x4_f32(
                 false, a0, false, b0, (short)0, c0, false, false);
        c1 = __builtin_amdgcn_wmma_f32_16x16x4_f32(
                 false, a1, false, b1, (short)0, c1, false, false);
    }

    // Tail: up to two K=4 chunks, branchless clamped indices (A entry forced
    // to zero for out-of-range edges; clamped load stays in-bounds & finite).
    for (; e < end; e += 4) {
        const int ea = e + khalf * 2;
        const int eb = ea + 1;
        const int ea_c = (ea < end) ? ea : start;
        const int eb_c = (eb < end) ? eb : start;

        const int sa = seg[ea_c], sb = seg[eb_c];
        const int ra = nidx[ea_c], rb = nidx[eb_c];

        v2f a; a.x = ((ea < end) & (sa == target)) ? 1.0f : 0.0f;
               a.y = ((eb < end) & (sb == target)) ? 1.0f : 0.0f;
        v2f b; b.x = feat[ra * 64 + col0 + m];
               b.y = feat[rb * 64 + col0 + m];

        c0 = __builtin_amdgcn_wmma_f32_16x16x4_f32(
                 false, a, false, b, (short)0, c0, false, false);
    }

    c0 += c1;   // merge the two accumulator chains (elementwise f32 adds)

    // C/D layout: VGPR r -> M = r + 8*khalf, N = m. Each 16-lane half stores a
    // contiguous 64B run per r.
    const int row0 = base + khalf * 8;
#pragma unroll
    for (int r = 0; r < 8; ++r) {
        const int row = row0 + r;
        if (row < n_nodes) {
            out[row * 64 + col0 + m] = c0[r];
        }
    }
}

extern "C" void kernel_launch(void* const* d_in, const int* in_sizes, int n_in,
                              void* d_out, int out_size, void* d_ws, size_t ws_size,
                              hipStream_t stream) {
    const float* feat = (const float*)d_in[0];
    const int*   nidx = (const int*)d_in[1];
    const int*   seg  = (const int*)d_in[2];
    float*       out  = (float*)d_out;

    const int n_nodes = in_sizes[0] / 64;
    const int n_edges = in_sizes[1];

    const int tiles  = (n_nodes + 15) / 16;
    const int waves  = tiles * 4;                 // 4 feature tiles per node tile
    const int threads = 256;                      // 8 waves per block (wave32)
    const int blocks  = (waves * 32 + threads - 1) / threads;

    seg_sum_wmma_kernel<<<blocks, threads, 0, stream>>>(feat, nidx, seg, out,
                                                        n_nodes, n_edges);
}